// FNO2DTime_18494129177291
// MI455X (gfx1250) — compile-verified
//
#include <hip/hip_runtime.h>

// ---------------------------------------------------------------------------
// FNO2DTime on MI455X (gfx1250), fp32 WMMA implementation.
//
// Truncated spectral conv (only 16x16x8 retained modes) computed as a chain
// of dense GEMMs (direct truncated DFTs) on V_WMMA_F32_16X16X4_F32.
// Per-wave 32x32 register blocking (4 accumulators) + LDS-staged transposed
// B panels (single ds_load_b64 per B fragment, bank-conflict-free padding).
// ---------------------------------------------------------------------------

typedef __attribute__((ext_vector_type(2))) float v2f;
typedef __attribute__((ext_vector_type(8))) float v8f;

#if __has_builtin(__builtin_amdgcn_wmma_f32_16x16x4_f32)
#define USE_WMMA_F32 1
#else
#define USE_WMMA_F32 0
#endif

// ---- problem constants ----
#define NB 4
#define NC 20           // WIDTH
#define NX 128
#define NY 128
#define NZ 20           // T
#define KXY 16          // retained kx/ky modes (0..7 and 120..127)
#define KZM 8           // retained kz modes
#define S_TOT 327680    // NX*NY*NZ

// ---- workspace layout (float offsets) ----
#define OFF_H    0ULL
#define OFF_F1   26214400ULL      // [bc][x][y][kz*2+ri]           20,971,520
#define OFF_F2   47185920ULL      // [bc][x][kz][ky'*2+ri]          2,621,440
#define OFF_F3   49807360ULL      // [bc][ky'][kz][kx'*2+ri]          327,680
#define OFF_F4   50135040ULL      // O4, same layout as F3            327,680
#define OFF_SUP  50462720ULL      // [bc][x][y][z]                 26,214,400
#define OFF_WZF  76677120ULL      // [20][16]                             320
#define OFF_WYF  76677440ULL      // [256][32]                           8192
#define OFF_WYI  76685632ULL      // [32][256]                           8192
#define OFF_WZI  76693824ULL      // [16][32] (cols 20..31 zero)          512

#if USE_WMMA_F32
__device__ __forceinline__ v8f wmma4(v2f a, v2f b, v8f c) {
    return __builtin_amdgcn_wmma_f32_16x16x4_f32(false, a, false, b,
                                                 (short)0, c, false, false);
}
#endif

// ===========================================================================
// Generic WMMA f32 GEMM core.
//   - one wave computes a 32 x (16*NT) output tile (2 M-subtiles, NT N-subtiles)
//   - B panel (K x N, row-major, ld=ldb) staged to LDS transposed [n][k] with
//     +2 row padding: 8B-aligned ds_load_b64 fragments, bank-conflict free.
// Fragment layouts per CDNA5 ISA 7.12.2:
//   A 16x4: lanes 0-15 rows, half0 K={k0,k0+1}, half1 K={k0+2,k0+3}
//   B 4x16: lane = N column, half0 K={k0,k0+1}, half1 K={k0+2,k0+3}
//   C/D:    half0 -> rows 0..7, half1 -> rows 8..15, col = lane&15
// ===========================================================================
template <int NT, int K, class AF, class SF>
__global__ __launch_bounds__(256) void k_gemm(int ntgroups, int nwaves, int N,
                                              AF A, const float* __restrict__ Bp,
                                              int ldb, SF St) {
    constexpr int Kp = K + 2;
    __shared__ __align__(16) float sB[8704];

    // cooperative transposed staging of the whole B panel (coalesced reads)
    for (int idx = threadIdx.x; idx < N * K; idx += 256) {
        int k = idx / N, n = idx - k * N;
        sB[n * Kp + k] = Bp[k * ldb + n];
    }
    __syncthreads();

    int w = blockIdx.x * 8 + (threadIdx.x >> 5);
    if (w < nwaves) {
        int mt = w / ntgroups, ng = w - mt * ntgroups;
        int m0 = mt << 5;                 // 32 rows per wave
        int n0 = ng * (NT * 16);
        int lane = threadIdx.x & 31;
        int half = lane >> 4, lm = lane & 15;

        v8f acc[NT][2];
#pragma unroll
        for (int j = 0; j < NT; ++j) { acc[j][0] = (v8f){}; acc[j][1] = (v8f){}; }

        for (int k0 = 0; k0 < K; k0 += 4) {
#if USE_WMMA_F32
            int ka = k0 + half * 2;
            v2f a0, a1;
            a0.x = A(m0 + lm, ka);      a0.y = A(m0 + lm, ka + 1);
            a1.x = A(m0 + 16 + lm, ka); a1.y = A(m0 + 16 + lm, ka + 1);
#pragma unroll
            for (int j = 0; j < NT; ++j) {
                v2f b = *(const v2f*)&sB[(n0 + j * 16 + lm) * Kp + ka];
                acc[j][0] = wmma4(a0, b, acc[j][0]);
                acc[j][1] = wmma4(a1, b, acc[j][1]);
            }
#else
            for (int kk = 0; kk < 4; ++kk) {
#pragma unroll
                for (int j = 0; j < NT; ++j) {
                    float bv = sB[(n0 + j * 16 + lm) * Kp + k0 + kk];
                    for (int r = 0; r < 8; ++r) {
                        acc[j][0][r] += A(m0 + half * 8 + r, k0 + kk) * bv;
                        acc[j][1][r] += A(m0 + 16 + half * 8 + r, k0 + kk) * bv;
                    }
                }
            }
#endif
        }
#pragma unroll
        for (int j = 0; j < NT; ++j)
            for (int mi = 0; mi < 2; ++mi)
                for (int r = 0; r < 8; ++r)
                    St(m0 + mi * 16 + half * 8 + r, n0 + j * 16 + lm, acc[j][mi][r]);
    }
}

// ---- load/store functors -------------------------------------------------
struct StRow { float* p; int ld;
    __device__ void operator()(int r, int c, float v) const { p[r * ld + c] = v; } };

// Stage A (Z fwd DFT): rows (b,c,x,y), K = z (contiguous in h)
struct LdA_A { const float* h;
    __device__ float operator()(int m, int k) const { return h[m * NZ + k]; } };

// Stage B (Y fwd): rows (bc,x,kz), k = ri*128 + y, reads F1[bc][x][y][kz*2+ri]
struct LdA_B { const float* a1;
    __device__ float operator()(int m, int k) const {
        int kz = m & 7, mxy = m >> 3; int yy = k & 127, ri = k >> 7;
        return a1[(mxy * 128 + yy) * 16 + kz * 2 + ri]; } };

// Stage C (X fwd): rows (bc,ky',kz), k = ri*128 + x, reads F2[bc][x][kz][ky'*2+ri]
struct LdA_C { const float* a2;
    __device__ float operator()(int m, int k) const {
        int kz = m & 7, kyp = (m >> 3) & 15, bc = m >> 7;
        int xx = k & 127, ri = k >> 7;
        return a2[((bc * 128 + xx) * 8 + kz) * 32 + kyp * 2 + ri]; } };

// Stage C' (X inv): rows (bo,ky',kz), k = ri*16 + kx', reads O4 row-major
struct LdA_Ci { const float* o4;
    __device__ float operator()(int m, int k) const {
        int kxp = k & 15, ri = k >> 4;
        return o4[m * 32 + kxp * 2 + ri]; } };

// Stage B' (Y inv): rows (bo,x,kz), k = ri*16 + ky', reads F2'[bo][ky'][kz][x*2+ri]
struct LdA_Bi { const float* g2;
    __device__ float operator()(int m, int k) const {
        int kz = m & 7, xx = (m >> 3) & 127, bo = m >> 10;
        int kyp = k & 15, ri = k >> 4;
        return g2[((bo * 16 + kyp) * 8 + kz) * 256 + xx * 2 + ri]; } };

// Stage A' (Z inv): rows (bo,x,y), k = ri*8 + kz, reads F1'[bo,x][kz][y*2+ri]
struct LdA_Ai { const float* g1;
    __device__ float operator()(int m, int k) const {
        int yy = m & 127, box = m >> 7; int kz = k & 7, ri = k >> 3;
        return g1[(box * 8 + kz) * 256 + yy * 2 + ri]; } };

// Stage A' store: sup[bo][x*128+y][z], pad cols 20..31 dropped
struct StSup { float* sup;
    __device__ void operator()(int m, int n, float v) const {
        if (n < NZ) { int xy = m & 16383, bo = m >> 14;
            sup[(unsigned long long)bo * S_TOT + xy * NZ + n] = v; } } };

// ===========================================================================
// Twiddle / DFT matrix construction (on device, deterministic per call)
// ===========================================================================
__global__ void k_twiddles(float* wzf, float* wyf, float* wyi, float* wzi) {
    const float PI2 = 6.28318530717958647692f;
    int t = blockIdx.x * blockDim.x + threadIdx.x;
    if (t < 320) {                       // WZF [z=20][kz*2+ri=16], e^{-i}
        int z = t >> 4, n = t & 15, kz = n >> 1, ri = n & 1;
        float th = PI2 * (float)(kz * z) / 20.0f;
        wzf[t] = ri ? -__sinf(th) : __cosf(th);
        return;
    }
    t -= 320;
    if (t < 8192) {                      // WYF [k=ri*128+y][ky'*2+ri = 32], e^{-i}
        int k = t >> 5, n = t & 31;
        int riK = k >> 7, yy = k & 127;
        int kyp = n >> 1, riN = n & 1;
        int ky = (kyp < 8) ? kyp : (112 + kyp);
        float th = PI2 * (float)(ky * yy) / 128.0f;
        float c = __cosf(th), s = __sinf(th);
        wyf[t] = riK ? (riN ? c : s) : (riN ? -s : c);
        return;
    }
    t -= 8192;
    if (t < 8192) {                      // WYI [k=ri*16+ky'][y*2+ri = 256], e^{+i}
        int k = t >> 8, n = t & 255;
        int riK = k >> 4; int kyp = k & 15;
        int yy = n >> 1, riN = n & 1;
        int ky = (kyp < 8) ? kyp : (112 + kyp);
        float th = PI2 * (float)(ky * yy) / 128.0f;
        float c = __cosf(th), s = __sinf(th);
        wyi[t] = riK ? (riN ? c : -s) : (riN ? s : c);
        return;
    }
    t -= 8192;
    if (t < 512) {                       // WZI [k=ri*8+kz][z pad 32], irfft + 1/(XYZ)
        int k = t >> 5, n = t & 31;
        int riK = k >> 3, kz = k & 7;
        if (n < NZ) {
            float ck = ((kz == 0) ? 1.0f : 2.0f) * (1.0f / 327680.0f);
            float th = PI2 * (float)(kz * n) / 20.0f;
            wzi[t] = riK ? (-ck * __sinf(th)) : (ck * __cosf(th));
        } else wzi[t] = 0.0f;
    }
}

// ===========================================================================
// fc0: h[b,c,x,y,z] = bias + gx*w0 + gy*w1 + gt(z)*w2 + sum_j x_j * w[3+j]
// (x-dependent partial is z-invariant -> computed once per (b,x,y))
// ===========================================================================
__global__ void k_fc0(const float* __restrict__ x, const float* __restrict__ w,
                      const float* __restrict__ bias, float* __restrict__ h) {
    int t = blockIdx.x * blockDim.x + threadIdx.x;
    if (t >= NB * NX * NY) return;
    int y = t & 127, xx = (t >> 7) & 127, b = t >> 14;
    const float* xin = x + (unsigned long long)t * NZ;
    float xv[NC];
    for (int j = 0; j < NC; ++j) xv[j] = xin[j];
    float gx = (float)xx * (1.0f / 127.0f);
    float gy = (float)y * (1.0f / 127.0f);
    int sbase = (xx * 128 + y) * NZ;
    for (int c = 0; c < NC; ++c) {
        float acc = bias[c] + gx * w[0 * NC + c] + gy * w[1 * NC + c];
        for (int j = 0; j < NC; ++j) acc += xv[j] * w[(3 + j) * NC + c];
        float wt = w[2 * NC + c];
        float* hp = h + ((unsigned long long)(b * NC + c)) * S_TOT + sbase;
        for (int z = 0; z < NZ; ++z)
            hp[z] = acc + ((float)(z + 1) * (1.0f / 20.0f)) * wt;
    }
}

// ===========================================================================
// Per-mode complex channel mix: O[b,o,mode] = sum_i H[b,i,mode] * W[blk,i,o,mode]
// ===========================================================================
__global__ void k_specmul(const float* __restrict__ a3, const float* __restrict__ wr,
                          const float* __restrict__ wi, float* __restrict__ o4) {
    int t = blockIdx.x * blockDim.x + threadIdx.x;
    if (t >= KXY * KZM * KXY * NB * NC) return;   // 163,840
    int o = t % NC; int r = t / NC;
    int b = r & 3; r >>= 2;
    int kxp = r & 15; r >>= 4;
    int kz = r & 7; int kyp = r >> 3;
    int blk = ((kxp >= 8) ? 1 : 0) + ((kyp >= 8) ? 2 : 0);
    int kxm = kxp & 7, kym = kyp & 7;
    float Or = 0.0f, Oi = 0.0f;
    for (int i = 0; i < NC; ++i) {
        unsigned long long hrow = (((unsigned long long)(b * NC + i) * 16 + kyp) * 8 + kz) * 32
                                  + kxp * 2;
        float Hr = a3[hrow], Hi = a3[hrow + 1];
        unsigned long long widx = ((((unsigned long long)(blk * NC + i) * NC + o) * 8 + kxm) * 8
                                   + kym) * 8 + kz;
        float Wr = wr[widx], Wi = wi[widx];
        Or += Hr * Wr - Hi * Wi;
        Oi += Hr * Wi + Hi * Wr;
    }
    unsigned long long orow = (((unsigned long long)(b * NC + o) * 16 + kyp) * 8 + kz) * 32
                              + kxp * 2;
    o4[orow] = Or;
    o4[orow + 1] = Oi;
}

// ===========================================================================
// combine: h = relu(sup + conv_w @ h + conv_b), in place on h (per point)
// ===========================================================================
__global__ void k_combine(const float* __restrict__ sup, const float* __restrict__ cw,
                          const float* __restrict__ cb, float* __restrict__ h) {
    int p = blockIdx.x * blockDim.x + threadIdx.x;
    if (p >= NB * S_TOT) return;
    int b = p / S_TOT; int s = p - b * S_TOT;
    float hv[NC];
    for (int i = 0; i < NC; ++i)
        hv[i] = h[(unsigned long long)(b * NC + i) * S_TOT + s];
    for (int o = 0; o < NC; ++o) {
        float acc = cb[o];
        for (int i = 0; i < NC; ++i) acc += cw[o * NC + i] * hv[i];
        acc += sup[(unsigned long long)(b * NC + o) * S_TOT + s];
        h[(unsigned long long)(b * NC + o) * S_TOT + s] = fmaxf(acc, 0.0f);
    }
}

// ===========================================================================
// fused fc1 (WMMA, [32 pts x 20] x [20 x 128]) + ReLU + fc2 dot in epilogue.
// fc1 weights staged to LDS transposed [n][k] (+2 pad).
// ===========================================================================
__global__ __launch_bounds__(256) void k_fc12(const float* __restrict__ h,
                                              const float* __restrict__ w1,
                                              const float* __restrict__ b1,
                                              const float* __restrict__ w2,
                                              const float* __restrict__ b2,
                                              float* __restrict__ out) {
    __shared__ __align__(16) float sW[128 * 22];
    for (int idx = threadIdx.x; idx < NC * 128; idx += 256) {
        int k = idx >> 7, n = idx & 127;
        sW[n * 22 + k] = w1[idx];
    }
    __syncthreads();

    int w = blockIdx.x * 8 + (threadIdx.x >> 5);
    if (w >= (NB * S_TOT) / 32) return;
    int m0 = w << 5;                              // 32 points per wave
    int b = m0 / S_TOT; int s0 = m0 - b * S_TOT;  // 32 | S_TOT
    int lane = threadIdx.x & 31;
    int half = lane >> 4, lm = lane & 15;
    float part0[8], part1[8];
    for (int r = 0; r < 8; ++r) { part0[r] = 0.0f; part1[r] = 0.0f; }

    for (int nt = 0; nt < 8; ++nt) {
        v8f acc0 = {}, acc1 = {};
        for (int k0 = 0; k0 < NC; k0 += 4) {
#if USE_WMMA_F32
            int ka = k0 + half * 2;
            const float* hb = h + (unsigned long long)(b * NC + ka) * S_TOT + s0 + lm;
            v2f a0, a1, bb;
            a0.x = hb[0];  a0.y = hb[S_TOT];
            a1.x = hb[16]; a1.y = hb[S_TOT + 16];
            bb = *(const v2f*)&sW[(nt * 16 + lm) * 22 + ka];
            acc0 = wmma4(a0, bb, acc0);
            acc1 = wmma4(a1, bb, acc1);
#else
            for (int kk = 0; kk < 4; ++kk) {
                float bv = sW[(nt * 16 + lm) * 22 + k0 + kk];
                for (int r = 0; r < 8; ++r) {
                    acc0[r] += h[(unsigned long long)(b * NC + k0 + kk) * S_TOT
                                 + s0 + half * 8 + r] * bv;
                    acc1[r] += h[(unsigned long long)(b * NC + k0 + kk) * S_TOT
                                 + s0 + 16 + half * 8 + r] * bv;
                }
            }
#endif
        }
        float bias = b1[nt * 16 + lm], wv = w2[nt * 16 + lm];
        for (int r = 0; r < 8; ++r) {
            part0[r] += fmaxf(acc0[r] + bias, 0.0f) * wv;
            part1[r] += fmaxf(acc1[r] + bias, 0.0f) * wv;
        }
    }
    // reduce over the 16 lanes of each half (hidden dim) -> row sums
    for (int r = 0; r < 8; ++r) {
        for (int off = 1; off < 16; off <<= 1) {
            part0[r] += __shfl_xor(part0[r], off, 32);
            part1[r] += __shfl_xor(part1[r], off, 32);
        }
    }
    if (lm == 0) {
        float bb2 = b2[0];
        for (int r = 0; r < 8; ++r) {
            out[m0 + half * 8 + r] = part0[r] + bb2;
            out[m0 + 16 + half * 8 + r] = part1[r] + bb2;
        }
    }
}

// ===========================================================================
// host side
// ===========================================================================
template <int NT, int K, class AF, class SF>
static void launch_gemm(int mtiles32, int ntgroups, int N, AF a,
                        const float* Bp, int ldb, SF s, hipStream_t st) {
    int waves = mtiles32 * ntgroups;
    int blocks = (waves + 7) / 8;
    k_gemm<NT, K, AF, SF><<<blocks, 256, 0, st>>>(ntgroups, waves, N, a, Bp, ldb, s);
}

extern "C" void kernel_launch(void* const* d_in, const int* in_sizes, int n_in,
                              void* d_out, int out_size, void* d_ws, size_t ws_size,
                              hipStream_t stream) {
    const float* x       = (const float*)d_in[0];
    const float* spec_wr = (const float*)d_in[1];
    const float* spec_wi = (const float*)d_in[2];
    const float* conv_w  = (const float*)d_in[3];
    const float* conv_b  = (const float*)d_in[4];
    const float* fc0_w   = (const float*)d_in[5];
    const float* fc0_b   = (const float*)d_in[6];
    const float* fc1_w   = (const float*)d_in[7];
    const float* fc1_b   = (const float*)d_in[8];
    const float* fc2_w   = (const float*)d_in[9];
    const float* fc2_b   = (const float*)d_in[10];
    float* out = (float*)d_out;

    float* ws  = (float*)d_ws;
    float* H   = ws + OFF_H;
    float* F1  = ws + OFF_F1;
    float* F2  = ws + OFF_F2;
    float* F3  = ws + OFF_F3;
    float* F4  = ws + OFF_F4;
    float* SUP = ws + OFF_SUP;
    float* WZF = ws + OFF_WZF;
    float* WYF = ws + OFF_WYF;
    float* WYI = ws + OFF_WYI;
    float* WZI = ws + OFF_WZI;

    // twiddle/DFT matrices (17,216 values)
    k_twiddles<<<68, 256, 0, stream>>>(WZF, WYF, WYI, WZI);

    // fc0 lift -> H [b][c][x][y][z]
    k_fc0<<<(NB * NX * NY + 255) / 256, 256, 0, stream>>>(x, fc0_w, fc0_b, H);

    for (int l = 0; l < 4; ++l) {
        // --- forward truncated DFTs ---
        // Z: [1,310,720 x 20] x [20 x 16] -> F1
        launch_gemm<1, 20>(40960, 1, 16, LdA_A{H}, WZF, 16, StRow{F1, 16}, stream);
        // Y: [81,920 x 256] x [256 x 32] -> F2
        launch_gemm<2, 256>(2560, 1, 32, LdA_B{F1}, WYF, 32, StRow{F2, 32}, stream);
        // X: [10,240 x 256] x [256 x 32] -> F3
        launch_gemm<2, 256>(320, 1, 32, LdA_C{F2}, WYF, 32, StRow{F3, 32}, stream);
        // --- per-mode complex channel mix ---
        k_specmul<<<640, 256, 0, stream>>>(F3,
            spec_wr + (unsigned long long)l * 4 * NC * NC * 512,
            spec_wi + (unsigned long long)l * 4 * NC * NC * 512, F4);
        // --- inverse DFTs ---
        // X inv: [10,240 x 32] x [32 x 256] -> F2 (reuse)
        launch_gemm<2, 32>(320, 8, 256, LdA_Ci{F4}, WYI, 256, StRow{F2, 256}, stream);
        // Y inv: [81,920 x 32] x [32 x 256] -> F1 (reuse)
        launch_gemm<2, 32>(2560, 8, 256, LdA_Bi{F2}, WYI, 256, StRow{F1, 256}, stream);
        // Z inv (real, + 1/XYZ): [1,310,720 x 16] x [16 x 20pad32] -> SUP
        launch_gemm<2, 16>(40960, 1, 32, LdA_Ai{F1}, WZI, 32, StSup{SUP}, stream);
        // --- h = relu(sup + conv(h) + bias), in place ---
        k_combine<<<(NB * S_TOT + 255) / 256, 256, 0, stream>>>(
            SUP, conv_w + l * NC * NC, conv_b + l * NC, H);
    }

    // fused fc1 (WMMA) + ReLU + fc2
    k_fc12<<<(40960 + 7) / 8, 256, 0, stream>>>(H, fc1_w, fc1_b, fc2_w, fc2_b, out);

    (void)in_sizes; (void)n_in; (void)out_size; (void)ws_size;
}